// NawrotDownsampler_88295937671414
// MI455X (gfx1250) — compile-verified
//
#include <hip/hip_runtime.h>
#include <math.h>

#define BB 8
#define LL 2048
#define DD 1024
#define SS (LL + 1)   // L+1 segment rows (null group at 0)

typedef float v2f __attribute__((ext_vector_type(2)));
typedef float v8f __attribute__((ext_vector_type(8)));

// ---------------------------------------------------------------------------
// K0: pack W1 (row-major D x D) into WMMA B-fragment order.
// W1p[((g*D) + n)*4 + j] = W1[(4g + j)*D + n]   (g = K-group of 4, n = column)
// After packing, a lane's B frag (K pair for its half) is ONE aligned b64 load.
// ---------------------------------------------------------------------------
__global__ __launch_bounds__(256)
void k0_pack(const float* __restrict__ W1, float* __restrict__ W1p)
{
    const int idx = blockIdx.x * 256 + threadIdx.x;   // 0 .. 256*1024-1
    const int g = idx >> 10;                          // K group (0..255)
    const int n = idx & 1023;                         // column
    float4 v;
    v.x = W1[(size_t)(4 * g + 0) * DD + n];
    v.y = W1[(size_t)(4 * g + 1) * DD + n];
    v.z = W1[(size_t)(4 * g + 2) * DD + n];
    v.w = W1[(size_t)(4 * g + 3) * DD + n];
    ((float4*)W1p)[idx] = v;
}

// ---------------------------------------------------------------------------
// K1: fused boundary predictor.
//   h = relu(x @ W1 + b1); logits = h @ W2 + b2
// One block = 32-token strip (2 M-tiles). 8 waves; wave w owns N columns
// [w*128, w*128+128) as 8 tiles. Each B frag feeds 2 WMMAs (both M-tiles):
// per K-step: 2 A b64 + 8 B b64 loads -> 16 v_wmma_f32_16x16x4_f32.
// Frag layouts:
//  A (16x4 f32): lanes 0-15 -> M=lane, K=k0,k0+1 ; lanes 16-31 -> K=k0+2,k0+3
//  B (4x16 f32): mirrored (lane%16 = N, lane/16 selects K pair) via W1p packing
//  C/D (16x16 f32): vgpr r, lanes 0-15 -> M=r ; lanes 16-31 -> M=r+8 ; N=lane%16
// ---------------------------------------------------------------------------
__global__ __launch_bounds__(256)
void k1_gemm_logits(const float* __restrict__ x, const float* __restrict__ W1p,
                    const float* __restrict__ b1, const float* __restrict__ W2,
                    const float* __restrict__ b2, float* __restrict__ logits)
{
    __shared__ float lsum[8][2][16];

    const int row0  = blockIdx.x * 32;          // flattened (b*L + l) strip base
    const int wave  = threadIdx.x >> 5;
    const int lane  = threadIdx.x & 31;
    const int l16   = lane & 15;
    const int khalf = (lane >> 4) << 1;         // 0 or 2
    const int n0w   = wave * 128;

    const float* xr0 = x + (size_t)(row0 + l16) * DD + khalf;
    const float* xr1 = xr0 + (size_t)16 * DD;
    const float* pB  = W1p + ((size_t)n0w + l16) * 4 + khalf;

    v8f acc[2][8] = {};

    for (int g = 0; g < DD / 4; ++g) {
        const v2f a0 = *(const v2f*)(xr0 + g * 4);
        const v2f a1 = *(const v2f*)(xr1 + g * 4);
        // WGP-scope prefetch of packed W1 16 K-groups (64 K) ahead
        __builtin_prefetch(pB + (size_t)16 * DD * 4, 0, 3);
#pragma unroll
        for (int t = 0; t < 8; ++t) {
            const v2f bf = *(const v2f*)(pB + t * 64);   // t*16 cols * 4 floats
            acc[0][t] = __builtin_amdgcn_wmma_f32_16x16x4_f32(
                false, a0, false, bf, (short)0, acc[0][t], false, false);
            acc[1][t] = __builtin_amdgcn_wmma_f32_16x16x4_f32(
                false, a1, false, bf, (short)0, acc[1][t], false, false);
        }
        pB += (size_t)DD * 4;
    }

    // Epilogue: bias + ReLU + dot with W2 (per-lane partials over wave's N slab)
    float plog[2][8];
#pragma unroll
    for (int m = 0; m < 2; ++m)
#pragma unroll
        for (int r = 0; r < 8; ++r) plog[m][r] = 0.f;

#pragma unroll
    for (int t = 0; t < 8; ++t) {
        const int n    = n0w + t * 16 + l16;
        const float bv = b1[n];
        const float wv = W2[n];
#pragma unroll
        for (int m = 0; m < 2; ++m)
#pragma unroll
            for (int r = 0; r < 8; ++r) {
                float h = acc[m][t][r] + bv;
                h = h > 0.f ? h : 0.f;
                plog[m][r] += h * wv;
            }
    }
    // Reduce across the 16 lanes of each half-wave (they hold different N)
#pragma unroll
    for (int m = 0; m < 2; ++m)
#pragma unroll
        for (int r = 0; r < 8; ++r) {
            float v = plog[m][r];
            v += __shfl_xor(v, 1);
            v += __shfl_xor(v, 2);
            v += __shfl_xor(v, 4);
            v += __shfl_xor(v, 8);
            plog[m][r] = v;
        }
    // lane 0 -> M=0..7, lane 16 -> M=8..15 (per M-tile)
    if (l16 == 0) {
        const int mbase = (lane >> 4) * 8;
#pragma unroll
        for (int m = 0; m < 2; ++m)
#pragma unroll
            for (int r = 0; r < 8; ++r) lsum[wave][m][mbase + r] = plog[m][r];
    }
    __syncthreads();
    // Deterministic ordered cross-wave sum
    if (threadIdx.x < 32) {
        const int m = threadIdx.x >> 4;
        const int i = threadIdx.x & 15;
        float s = b2[0];
#pragma unroll
        for (int w = 0; w < 8; ++w) s += lsum[w][m][i];
        logits[row0 + m * 16 + i] = s;
    }
}

// ---------------------------------------------------------------------------
// K2: per-batch boundary sample + inclusive scan + segment ranges.
// One block per batch, 1024 threads (2 tokens each).
// ---------------------------------------------------------------------------
__global__ __launch_bounds__(1024)
void k2_scan(const float* __restrict__ logits, const float* __restrict__ noise,
             int* __restrict__ segIncl, int* __restrict__ segCnt,
             int* __restrict__ segSt)
{
    __shared__ int ps[1024];
    __shared__ int cnt[SS];
    __shared__ int stt[SS];

    const int b = blockIdx.x;
    const int t = threadIdx.x;

    for (int i = t; i < SS; i += 1024) { cnt[i] = 0; stt[i] = 0x7FFFFFFF; }

    const int l0 = 2 * t;
    int h0, h1;
    {
        const float nz   = noise[b * LL + l0];
        const float z    = logits[b * LL + l0] + logf(nz) - log1pf(-nz);
        const float soft = 1.f / (1.f + expf(-z));   // TEMP = 1.0
        h0 = soft > 0.5f ? 1 : 0;                    // THRESH = 0.5
    }
    {
        const float nz   = noise[b * LL + l0 + 1];
        const float z    = logits[b * LL + l0 + 1] + logf(nz) - log1pf(-nz);
        const float soft = 1.f / (1.f + expf(-z));
        h1 = soft > 0.5f ? 1 : 0;
    }
    ps[t] = h0 + h1;
    __syncthreads();
    // Hillis-Steele inclusive scan over 1024 pair-sums
    for (int off = 1; off < 1024; off <<= 1) {
        const int add = (t >= off) ? ps[t - off] : 0;
        __syncthreads();
        ps[t] += add;
        __syncthreads();
    }
    const int excl  = ps[t] - (h0 + h1);   // boundaries strictly before token l0
    const int incl0 = excl + h0;
    const int incl1 = incl0 + h1;
    segIncl[b * LL + l0]     = incl0;
    segIncl[b * LL + l0 + 1] = incl1;
    // token l0 belongs to down-segment 'excl'; token l0+1 to 'incl0'
    atomicAdd(&cnt[excl], 1);
    atomicMin(&stt[excl], l0);
    atomicAdd(&cnt[incl0], 1);
    atomicMin(&stt[incl0], l0 + 1);
    __syncthreads();
    for (int i = t; i < SS; i += 1024) {
        segCnt[b * SS + i] = cnt[i];
        segSt [b * SS + i] = stt[i];
    }
}

// ---------------------------------------------------------------------------
// K3: segment means into shortBuf (B x (L+1) x D); row 0 = null_group.
// Segments are contiguous token ranges [st, st+cnt).
// ---------------------------------------------------------------------------
__global__ __launch_bounds__(256)
void k3_down(const float* __restrict__ x, const float* __restrict__ null_group,
             const int* __restrict__ segCnt, const int* __restrict__ segSt,
             float* __restrict__ shortBuf)
{
    const int b = blockIdx.y;
    const int s = blockIdx.x;                    // 0..LL
    const int t = threadIdx.x;                   // 256 threads x 4 floats = D
    float* orow = shortBuf + ((size_t)b * SS + s) * DD;

    if (s == 0) {
        const float4 v = *(const float4*)(null_group + t * 4);
        *(float4*)(orow + t * 4) = v;
        return;
    }
    const int k    = s - 1;
    const int cntv = segCnt[b * SS + k];
    if (cntv == 0) return;                       // never referenced by K4
    const int st0  = segSt[b * SS + k];
    const float inv = 1.0f / ((float)cntv + 1e-9f);

    float s0 = 0.f, s1 = 0.f, s2 = 0.f, s3 = 0.f;
    const float* xr = x + ((size_t)b * LL + st0) * DD + t * 4;
    for (int r = 0; r < cntv; ++r) {
        const float4 v = *(const float4*)xr;
        s0 += v.x; s1 += v.y; s2 += v.z; s3 += v.w;
        xr += DD;
    }
    float4 o; o.x = s0 * inv; o.y = s1 * inv; o.z = s2 * inv; o.w = s3 * inv;
    *(float4*)(orow + t * 4) = o;
}

// ---------------------------------------------------------------------------
// K4: up[l] = shortBuf[seg_incl[l]] * 1/(1+1e-9)
// ---------------------------------------------------------------------------
__global__ __launch_bounds__(256)
void k4_up(const float* __restrict__ shortBuf, const int* __restrict__ segIncl,
           float* __restrict__ out)
{
    const int b = blockIdx.y;
    const int l = blockIdx.x;
    const int s = segIncl[b * LL + l];
    const float sc = 1.0f / (1.0f + 1e-9f);
    const float4 v =
        *(const float4*)(shortBuf + ((size_t)b * SS + s) * DD + threadIdx.x * 4);
    float4 o; o.x = v.x * sc; o.y = v.y * sc; o.z = v.z * sc; o.w = v.w * sc;
    *(float4*)(out + ((size_t)b * LL + l) * DD + threadIdx.x * 4) = o;
}

// ---------------------------------------------------------------------------
extern "C" void kernel_launch(void* const* d_in, const int* in_sizes, int n_in,
                              void* d_out, int out_size, void* d_ws, size_t ws_size,
                              hipStream_t stream)
{
    const float* x     = (const float*)d_in[0];
    const float* noise = (const float*)d_in[1];
    const float* W1    = (const float*)d_in[2];
    const float* b1    = (const float*)d_in[3];
    const float* W2    = (const float*)d_in[4];
    const float* b2    = (const float*)d_in[5];
    const float* ng    = (const float*)d_in[6];
    float* out = (float*)d_out;

    // Workspace: W1p | logits | segIncl | segCnt | segSt | shortBuf (~68.3 MiB)
    char* ws = (char*)d_ws;
    float* W1p      = (float*)ws;  ws += (size_t)DD * DD * sizeof(float);
    float* logits   = (float*)ws;  ws += (size_t)BB * LL * sizeof(float);
    int*   segIncl  = (int*)ws;    ws += (size_t)BB * LL * sizeof(int);
    int*   segCnt   = (int*)ws;    ws += (size_t)BB * SS * sizeof(int);
    int*   segSt    = (int*)ws;    ws += (size_t)BB * SS * sizeof(int);
    float* shortBuf = (float*)ws;  // BB * SS * DD floats (~64 MiB)

    k0_pack<<<DD * DD / 4 / 256, 256, 0, stream>>>(W1, W1p);
    k1_gemm_logits<<<BB * LL / 32, 256, 0, stream>>>(x, W1p, b1, W2, b2, logits);
    k2_scan<<<BB, 1024, 0, stream>>>(logits, noise, segIncl, segCnt, segSt);
    k3_down<<<dim3(SS, BB), 256, 0, stream>>>(x, ng, segCnt, segSt, shortBuf);
    k4_up<<<dim3(LL, BB), 256, 0, stream>>>(shortBuf, segIncl, out);
}